// HierarchicalSoftmax_28733331210339
// MI455X (gfx1250) — compile-verified
//
#include <hip/hip_runtime.h>
#include <hip/hip_bf16.h>
#include <math.h>
#include <stdint.h>

#define NHID      1024
#define NCLASSES  224
#define NTOK      225
#define BATCH     2048
#define NTILE_TOP 14             // 224 / 16
#define NTILE_BOT 15             // ceil(225/16)
#define NTOK_PAD  (NTILE_BOT*16) // 240
#define APAD      4              // 16B row pad -> conflict-free ds_load_b64

typedef __attribute__((ext_vector_type(2))) float v2f;
typedef __attribute__((ext_vector_type(8))) float v8f;

// ---- gfx1250 async global->LDS copy (16B per lane), tracked by ASYNCcnt ----
__device__ __forceinline__ void async_copy_f4(uint32_t lds_off, const float* gptr) {
    asm volatile("global_load_async_to_lds_b128 %0, %1, off"
                 :: "v"(lds_off), "v"(gptr) : "memory");
}
__device__ __forceinline__ void async_wait0() {
    asm volatile("s_wait_asynccnt 0" ::: "memory");
}

// ---- wave32 reductions ----
__device__ __forceinline__ float wave_max(float v) {
#pragma unroll
    for (int m = 16; m; m >>= 1) v = fmaxf(v, __shfl_xor(v, m, 32));
    return v;
}
__device__ __forceinline__ float wave_sum(float v) {
#pragma unroll
    for (int m = 16; m; m >>= 1) v += __shfl_xor(v, m, 32);
    return v;
}

// ---------------- class binning ----------------
__global__ void zero_counts_kernel(int* __restrict__ cnt, int* __restrict__ cur) {
    int t = threadIdx.x;
    if (t < NCLASSES) { cnt[t] = 0; cur[t] = 0; }
}

__global__ void count_kernel(const int* __restrict__ labels, int* __restrict__ cnt) {
    int i = blockIdx.x * blockDim.x + threadIdx.x;
    if (i < BATCH) atomicAdd(&cnt[labels[i] / NTOK], 1);
}

__global__ void prefix_kernel(const int* __restrict__ cnt, int* __restrict__ off) {
    if (threadIdx.x == 0 && blockIdx.x == 0) {
        int acc = 0;
        for (int i = 0; i < NCLASSES; ++i) { off[i] = acc; acc += cnt[i]; }
        off[NCLASSES] = acc;
    }
}

__global__ void scatter_kernel(const int* __restrict__ labels, const int* __restrict__ off,
                               int* __restrict__ cur, int* __restrict__ sidx) {
    int i = blockIdx.x * blockDim.x + threadIdx.x;
    if (i < BATCH) {
        int c = labels[i] / NTOK;
        int p = atomicAdd(&cur[c], 1);
        sidx[off[c] + p] = i;
    }
}

// ---------------- top: [2048,1024] x [1024,224] + softmax(sel) ----------------
__global__ __launch_bounds__(NTILE_TOP * 32)
void top_kernel(const float* __restrict__ x, const float* __restrict__ Wt,
                const float* __restrict__ bt, const int* __restrict__ labels,
                float* __restrict__ psel) {
    __shared__ float lg[16][NCLASSES];

    const int wave = threadIdx.x >> 5;
    const int lane = threadIdx.x & 31;
    const int row0 = blockIdx.x << 4;
    const int n0   = wave << 4;
    const int hl   = lane & 15;
    const int kadd = (lane >> 4) << 1;  // lanes 16-31 carry K=2,3 of the quad

    const float* Ap = x  + (size_t)(row0 + hl) * NHID + kadd;
    const float* Bp = Wt + (size_t)kadd * NCLASSES + (n0 + hl);

    v8f acc0 = {}, acc1 = {};           // two chains: hide WMMA RAW latency
#pragma unroll 4
    for (int k0 = 0; k0 < NHID; k0 += 8) {
        v2f a0, b0, a1, b1;
        a0.x = Ap[k0];     a0.y = Ap[k0 + 1];
        b0.x = Bp[(size_t)k0 * NCLASSES];
        b0.y = Bp[(size_t)(k0 + 1) * NCLASSES];
        a1.x = Ap[k0 + 4]; a1.y = Ap[k0 + 5];
        b1.x = Bp[(size_t)(k0 + 4) * NCLASSES];
        b1.y = Bp[(size_t)(k0 + 5) * NCLASSES];
        acc0 = __builtin_amdgcn_wmma_f32_16x16x4_f32(false, a0, false, b0,
                                                     (short)0, acc0, false, false);
        acc1 = __builtin_amdgcn_wmma_f32_16x16x4_f32(false, a1, false, b1,
                                                     (short)0, acc1, false, false);
    }
    v8f acc = acc0 + acc1;

    const int   mBase = (lane < 16) ? 0 : 8;
    const int   nloc  = n0 + hl;
    const float bias  = bt[nloc];
#pragma unroll
    for (int j = 0; j < 8; ++j) lg[mBase + j][nloc] = acc[j] + bias;

    __syncthreads();

    // wave-parallel softmax: rows striped over waves, cols over lanes
    for (int r = wave; r < 16; r += NTILE_TOP) {
        float mx = -INFINITY;
        for (int t = lane; t < NCLASSES; t += 32) mx = fmaxf(mx, lg[r][t]);
        mx = wave_max(mx);
        float s = 0.f;
        for (int t = lane; t < NCLASSES; t += 32) s += __expf(lg[r][t] - mx);
        s = wave_sum(s);
        if (lane == 0) {
            const int c = labels[row0 + r] / NTOK;
            psel[row0 + r] = __expf(lg[r][c] - mx) / s;
        }
    }
}

// ---------------- bottom: per-class grouped GEMM + softmax ----------------
__global__ __launch_bounds__(NTILE_BOT * 32)
void bottom_kernel(const float* __restrict__ x, const float* __restrict__ Wb,
                   const float* __restrict__ bb, const int* __restrict__ labels,
                   const int* __restrict__ cnt, const int* __restrict__ off,
                   const int* __restrict__ sidx, const float* __restrict__ psel,
                   float* __restrict__ out) {
    __shared__ __align__(16) float As[16][NHID + APAD];  // staged A tile (64KB+pad)
    __shared__ float lg[16][NTOK_PAD];
    __shared__ int   rows[16];

    const int c = blockIdx.x;
    const int n = cnt[c];
    if (n == 0) return;
    const int base = off[c];

    const int wave = threadIdx.x >> 5;
    const int lane = threadIdx.x & 31;
    const int hl   = lane & 15;
    const int kadd = (lane >> 4) << 1;
    const int n0   = wave << 4;
    const int nloc = n0 + hl;
    const int ncl  = (nloc < NTOK - 1) ? nloc : (NTOK - 1);

    const float*   Bp     = Wb + (size_t)c * NHID * NTOK + (size_t)kadd * NTOK + ncl;
    const uint32_t asBase = (uint32_t)(uintptr_t)&As[0][0];

    for (int chunk = 0; chunk < n; chunk += 16) {
        if (threadIdx.x < 16) {
            int s = chunk + threadIdx.x;
            rows[threadIdx.x] = sidx[base + (s < n ? s : 0)];
        }
        __syncthreads();

        // async-stage the gathered 16x1024 A tile into LDS (16B per lane)
        for (int e = threadIdx.x; e < 16 * (NHID / 4); e += NTILE_BOT * 32) {
            const int m  = e >> 8;       // row 0..15
            const int c4 = e & 255;      // float4 index within row
            const float* g = x + (size_t)rows[m] * NHID + (c4 << 2);
            const uint32_t l = asBase + (uint32_t)((m * (NHID + APAD) + (c4 << 2)) << 2);
            async_copy_f4(l, g);
        }
        async_wait0();
        __syncthreads();

        const float* Arow = &As[hl][kadd];

        v8f acc0 = {}, acc1 = {};
#pragma unroll 4
        for (int k0 = 0; k0 < NHID; k0 += 8) {
            v2f a0, b0, a1, b1;
            a0.x = Arow[k0];     a0.y = Arow[k0 + 1];
            b0.x = Bp[(size_t)k0 * NTOK];
            b0.y = Bp[(size_t)(k0 + 1) * NTOK];
            a1.x = Arow[k0 + 4]; a1.y = Arow[k0 + 5];
            b1.x = Bp[(size_t)(k0 + 4) * NTOK];
            b1.y = Bp[(size_t)(k0 + 5) * NTOK];
            acc0 = __builtin_amdgcn_wmma_f32_16x16x4_f32(false, a0, false, b0,
                                                         (short)0, acc0, false, false);
            acc1 = __builtin_amdgcn_wmma_f32_16x16x4_f32(false, a1, false, b1,
                                                         (short)0, acc1, false, false);
        }
        v8f acc = acc0 + acc1;

        const int   mBase = (lane < 16) ? 0 : 8;
        const float bias  = (nloc < NTOK) ? bb[c * NTOK + nloc] : 0.f;
#pragma unroll
        for (int j = 0; j < 8; ++j) lg[mBase + j][nloc] = acc[j] + bias;

        __syncthreads();

        // wave-parallel softmax over 225 logits per in-range row
        for (int r = wave; r < 16; r += NTILE_BOT) {
            if (chunk + r < n) {
                float mx = -INFINITY;
                for (int t = lane; t < NTOK; t += 32) mx = fmaxf(mx, lg[r][t]);
                mx = wave_max(mx);
                float s = 0.f;
                for (int t = lane; t < NTOK; t += 32) s += __expf(lg[r][t] - mx);
                s = wave_sum(s);
                if (lane == 0) {
                    const int i  = rows[r];
                    const int tt = labels[i] % NTOK;
                    out[i] = (__expf(lg[r][tt] - mx) / s) * psel[i];
                }
            }
        }
        __syncthreads(); // protect rows[]/As/lg before next chunk
    }
}

// ---------------- launch ----------------
extern "C" void kernel_launch(void* const* d_in, const int* in_sizes, int n_in,
                              void* d_out, int out_size, void* d_ws, size_t ws_size,
                              hipStream_t stream) {
    (void)in_sizes; (void)n_in; (void)out_size; (void)ws_size;

    const float* x      = (const float*)d_in[0];
    const int*   labels = (const int*)d_in[1];
    const float* Wt     = (const float*)d_in[2];
    const float* bt     = (const float*)d_in[3];
    const float* Wb     = (const float*)d_in[4];
    const float* bbias  = (const float*)d_in[5];
    float*       out    = (float*)d_out;

    float* psel = (float*)d_ws;               // [BATCH]
    int*   cnt  = (int*)(psel + BATCH);       // [NCLASSES]
    int*   offp = cnt + NCLASSES;             // [NCLASSES+1]
    int*   cur  = offp + NCLASSES + 1;        // [NCLASSES]
    int*   sidx = cur + NCLASSES;             // [BATCH]

    zero_counts_kernel<<<1, 256, 0, stream>>>(cnt, cur);
    count_kernel<<<BATCH / 256, 256, 0, stream>>>(labels, cnt);
    prefix_kernel<<<1, 32, 0, stream>>>(cnt, offp);
    scatter_kernel<<<BATCH / 256, 256, 0, stream>>>(labels, offp, cur, sidx);

    top_kernel<<<BATCH / 16, NTILE_TOP * 32, 0, stream>>>(x, Wt, bt, labels, psel);
    bottom_kernel<<<NCLASSES, NTILE_BOT * 32, 0, stream>>>(x, Wb, bbias, labels,
                                                           cnt, offp, sidx, psel, out);
}